// GATE_Mechanism_VGAE_81819126989061
// MI455X (gfx1250) — compile-verified
//
#include <hip/hip_runtime.h>
#include <math.h>

typedef float v2f __attribute__((ext_vector_type(2)));
typedef float v8f __attribute__((ext_vector_type(8)));

#define NROWS 8192
#define KDIM  256
#define SEP   64

// ---- order-preserving float <-> uint mapping for deterministic atomic min/max ----
__device__ __forceinline__ unsigned int fmap_ord(float f) {
  unsigned int u = __float_as_uint(f);
  return (u & 0x80000000u) ? ~u : (u | 0x80000000u);
}
__device__ __forceinline__ float funmap_ord(unsigned int u) {
  u = (u & 0x80000000u) ? (u ^ 0x80000000u) : ~u;
  return __uint_as_float(u);
}

// ---- fp32 WMMA GEMM tile: one wave computes a 64x64 region of M = cur @ prev^T ----
// A (16x4 f32) lane layout: lanes 0-15 hold M=0..15; VGPR0/1 = K pair; lane halves
// split K {0,1} vs {2,3}.  B (4x16) mirrors it with N across lanes, and since
// B = prev^T, the B fragment load from row-major `prev` is the same pattern as A.
// 16 WMMAs per k-step fed by 8 b64 loads -> 0.5 vmem per wmma.
__device__ __forceinline__ void gemm_tile(const float* __restrict__ cur,
                                          const float* __restrict__ prev,
                                          int rowBase, int colBase, int lane,
                                          v8f acc[4][4]) {
  const int m    = lane & 15;
  const int koff = (lane >> 4) << 1;
  const float* pa = cur  + (size_t)(rowBase + m) * KDIM + koff;
  const float* pb = prev + (size_t)(colBase + m) * KDIM + koff;

#pragma unroll 2
  for (int k = 0; k < KDIM; k += 4) {
    v2f a[4];
    v2f b[4];
#pragma unroll
    for (int r = 0; r < 4; ++r)
      a[r] = *(const v2f*)(pa + (size_t)(r * 16) * KDIM + k);
#pragma unroll
    for (int c = 0; c < 4; ++c)
      b[c] = *(const v2f*)(pb + (size_t)(c * 16) * KDIM + k);
#pragma unroll
    for (int r = 0; r < 4; ++r) {
#pragma unroll
      for (int c = 0; c < 4; ++c) {
        acc[r][c] = __builtin_amdgcn_wmma_f32_16x16x4_f32(
            false, a[r], false, b[c], (short)0, acc[r][c], false, false);
      }
    }
  }
}

// ---- ws layout: ws[0]=mapped-min, ws[1]=mapped-max, ws[2..65]=hist counts ----
__global__ void init_ws_kernel(unsigned int* ws) {
  int t = threadIdx.x;
  if (t == 0) { ws[0] = 0xFFFFFFFFu; ws[1] = 0u; }
  if (t < SEP) ws[2 + t] = 0u;
}

__global__ __launch_bounds__(256)
void gemm_minmax_kernel(const float* __restrict__ cur,
                        const float* __restrict__ prev,
                        unsigned int* __restrict__ mm) {
  __shared__ unsigned int smin[8];
  __shared__ unsigned int smax[8];
  const int lane = threadIdx.x & 31;
  const int wave = threadIdx.x >> 5;
  const int rowBase = blockIdx.y * 128 + (wave >> 2) * 64;
  const int colBase = blockIdx.x * 256 + (wave & 3) * 64;

  v8f acc[4][4];
#pragma unroll
  for (int r = 0; r < 4; ++r)
#pragma unroll
    for (int c = 0; c < 4; ++c)
#pragma unroll
      for (int i = 0; i < 8; ++i) acc[r][c][i] = 0.0f;

  gemm_tile(cur, prev, rowBase, colBase, lane, acc);

  float mn = 3.402823466e38f, mx = -3.402823466e38f;
#pragma unroll
  for (int r = 0; r < 4; ++r)
#pragma unroll
    for (int c = 0; c < 4; ++c)
#pragma unroll
      for (int i = 0; i < 8; ++i) {
        float x = acc[r][c][i];
        mn = fminf(mn, x);
        mx = fmaxf(mx, x);
      }
  // wave32 reduction
#pragma unroll
  for (int off = 16; off >= 1; off >>= 1) {
    mn = fminf(mn, __shfl_xor(mn, off, 32));
    mx = fmaxf(mx, __shfl_xor(mx, off, 32));
  }
  if (lane == 0) { smin[wave] = fmap_ord(mn); smax[wave] = fmap_ord(mx); }
  __syncthreads();
  if (threadIdx.x == 0) {
    unsigned int u = smin[0], v = smax[0];
#pragma unroll
    for (int w = 1; w < 8; ++w) {
      u = min(u, smin[w]);
      v = max(v, smax[w]);
    }
    atomicMin(&mm[0], u);
    atomicMax(&mm[1], v);
  }
}

__global__ __launch_bounds__(256)
void gemm_hist_kernel(const float* __restrict__ cur,
                      const float* __restrict__ prev,
                      const unsigned int* __restrict__ mm,
                      unsigned int* __restrict__ ghist) {
  __shared__ unsigned int hist[SEP];
  if (threadIdx.x < SEP) hist[threadIdx.x] = 0u;
  __syncthreads();

  const int lane = threadIdx.x & 31;
  const int wave = threadIdx.x >> 5;
  const int rowBase = blockIdx.y * 128 + (wave >> 2) * 64;
  const int colBase = blockIdx.x * 256 + (wave & 3) * 64;

  v8f acc[4][4];
#pragma unroll
  for (int r = 0; r < 4; ++r)
#pragma unroll
    for (int c = 0; c < 4; ++c)
#pragma unroll
      for (int i = 0; i < 8; ++i) acc[r][c][i] = 0.0f;

  gemm_tile(cur, prev, rowBase, colBase, lane, acc);

  const float mn = funmap_ord(mm[0]);
  const float mx = funmap_ord(mm[1]);
  const float interval = (mx - mn) / (float)SEP;
  const float inv = (interval > 0.0f) ? (1.0f / interval) : 0.0f;

#pragma unroll
  for (int r = 0; r < 4; ++r)
#pragma unroll
    for (int c = 0; c < 4; ++c)
#pragma unroll
      for (int i = 0; i < 8; ++i) {
        float x = acc[r][c][i];
        int idx = (int)floorf((x - mn) * inv);
        idx = min(max(idx, 0), SEP - 1);
        atomicAdd(&hist[idx], 1u);  // ds_add_u32
      }
  __syncthreads();
  if (threadIdx.x < SEP) atomicAdd(&ghist[threadIdx.x], hist[threadIdx.x]);
}

__global__ void finalize_kernel(const unsigned int* __restrict__ ghist,
                                const float* __restrict__ W,
                                float* __restrict__ out) {
  if (threadIdx.x == 0) {
    float dot = 0.0f;
    for (int k = 0; k < SEP; ++k) {
      // beta = (sigmoid(hist) + sigmoid(hist)) / 2 = sigmoid(hist)
      float h = 1.0f / (1.0f + expf(-(float)ghist[k]));
      dot += h * W[k];
    }
    out[0] = 1.0f / (1.0f + expf(-dot));
  }
}

extern "C" void kernel_launch(void* const* d_in, const int* in_sizes, int n_in,
                              void* d_out, int out_size, void* d_ws, size_t ws_size,
                              hipStream_t stream) {
  const float* cur  = (const float*)d_in[0];  // [8192, 256]
  const float* prev = (const float*)d_in[1];  // [8192, 256]
  const float* W    = (const float*)d_in[2];  // [1, 64]
  float* out        = (float*)d_out;          // [1]
  unsigned int* ws  = (unsigned int*)d_ws;
  unsigned int* mm    = ws;       // [2]  mapped min / max
  unsigned int* ghist = ws + 2;   // [64] histogram counts

  init_ws_kernel<<<dim3(1), dim3(64), 0, stream>>>(ws);

  dim3 grid(NROWS / 256, NROWS / 128);  // 32 x 64 blocks of 256x128 tiles
  dim3 block(256);
  gemm_minmax_kernel<<<grid, block, 0, stream>>>(cur, prev, mm);
  gemm_hist_kernel<<<grid, block, 0, stream>>>(cur, prev, mm, ghist);
  finalize_kernel<<<dim3(1), dim3(64), 0, stream>>>(ghist, W, out);
}